// model_35064113004948
// MI455X (gfx1250) — compile-verified
//
#include <hip/hip_runtime.h>
#include <hip/hip_bf16.h>
#include <cstdint>

typedef float v2f __attribute__((ext_vector_type(2)));
typedef float v8f __attribute__((ext_vector_type(8)));

#define NEG_INF (-3.402823466e38f)
#define V8F_ZERO {0.f, 0.f, 0.f, 0.f, 0.f, 0.f, 0.f, 0.f}

// ---------------------------------------------------------------------------
// CDNA5 async global->LDS copy (ASYNCcnt-tracked). VDST = LDS byte address,
// VADDR = 64-bit global address (CDNA5 ISA). The low 32 bits
// of a flat pointer to __shared__ are the wave-relative LDS byte address
// (aperture mapping, CDNA5 ISA).
// ---------------------------------------------------------------------------
__device__ __forceinline__ void async_load_b128(const void* lds_dst, const void* gaddr) {
  uint32_t lds_addr = (uint32_t)(uintptr_t)lds_dst;
  asm volatile("global_load_async_to_lds_b128 %0, %1, off"
               :: "v"(lds_addr), "v"(gaddr)
               : "memory");
}
__device__ __forceinline__ void async_wait0() {
  asm volatile("s_wait_asynccnt 0x0" ::: "memory");
}

// ---------------------------------------------------------------------------
// Kernel 0: weight repacking into WMMA B-fragment order.
//   Edge weights (fold W_top-W_bot | W_bot):  Wep[nt][ks][lane][2]
//     element = Wcomb[k = ks*4 + 2*(lane>>4) + v][c = nt*16 + (lane&15)]
//   Conv weights: Wcp[ct][dk][ks(33)][lane][2]
//     element = Wc[co = ct*16+(lane&15)][ci = ks*4+2*(lane>>4)+v][dk], ci>=131 -> 0
// ---------------------------------------------------------------------------
__global__ void prep_weights(const float* __restrict__ We0, const float* __restrict__ We1,
                             const float* __restrict__ Wc0, const float* __restrict__ Wc1,
                             float* __restrict__ Wep0, float* __restrict__ Wep1,
                             float* __restrict__ Wcp0, float* __restrict__ Wcp1) {
  int tid = blockIdx.x * blockDim.x + threadIdx.x;
  int nthr = gridDim.x * blockDim.x;
  // edge0: K=64 -> ksteps=16, 16 N-tiles
  for (int i = tid; i < 16 * 16 * 64; i += nthr) {
    int v = i & 1, lane = (i >> 1) & 31;
    int rem = i >> 6, ks = rem & 15, nt = rem >> 4;
    int k = ks * 4 + 2 * (lane >> 4) + v;
    int c = nt * 16 + (lane & 15);
    Wep0[i] = (c < 128) ? (We0[k * 128 + c] - We0[(64 + k) * 128 + c])
                        : We0[(64 + k) * 128 + (c - 128)];
  }
  // edge1: K=128 -> ksteps=32, 16 N-tiles
  for (int i = tid; i < 16 * 32 * 64; i += nthr) {
    int v = i & 1, lane = (i >> 1) & 31;
    int rem = i >> 6, ks = rem & 31, nt = rem >> 5;
    int k = ks * 4 + 2 * (lane >> 4) + v;
    int c = nt * 16 + (lane & 15);
    Wep1[i] = (c < 128) ? (We1[k * 128 + c] - We1[(128 + k) * 128 + c])
                        : We1[(128 + k) * 128 + (c - 128)];
  }
  // conv: 8 co-tiles x 3 taps x 33 ksteps
  for (int i = tid; i < 8 * 3 * 33 * 64; i += nthr) {
    int v = i & 1, lane = (i >> 1) & 31;
    int rem = i >> 6, ks = rem % 33, rem2 = rem / 33;
    int dk = rem2 % 3, ct = rem2 / 3;
    int ci = ks * 4 + 2 * (lane >> 4) + v;
    int co = ct * 16 + (lane & 15);
    float a = 0.f, bb = 0.f;
    if (ci < 131) {
      a  = Wc0[(co * 131 + ci) * 3 + dk];
      bb = Wc1[(co * 131 + ci) * 3 + dk];
    }
    Wcp0[i] = a;
    Wcp1[i] = bb;
  }
}

// ---------------------------------------------------------------------------
// Kernel 1/3: fused EdgeConv (GEMM + group max-excluding-self + relu + ctx).
//   2x2 register blocking: 4 WMMAs per {2 A-frag ds_b64 + 2 B-frag global_b64}.
// ---------------------------------------------------------------------------
__global__ void edge_conv_kernel(const float* __restrict__ X,
                                 int bStride, int nStride, int tStride,
                                 int K, int Tlen,
                                 const float* __restrict__ Wp,   // packed [16][ksteps][32][2]
                                 const float* __restrict__ bias, // 128
                                 const float* __restrict__ ohs,  // (B*N) x 3
                                 float* __restrict__ S) {        // (B*N*Tlen) x 132
  extern __shared__ float smem[];
  float* Xs = smem;             // 128 * K
  float* As = Xs + 128 * K;     // 8*16*128
  float* Ys = As + 16384;       // 8*16*128

  const int b = blockIdx.y;
  const int t0 = blockIdx.x * 16;
  const int tid = threadIdx.x;
  const int lane = tid & 31;
  const int wave = tid >> 5;
  const int lh = lane >> 4;
  const int ll = lane & 15;

  // --- async stage X tile: row (n*16+m) = node n, time t0+m ---
  {
    int sn = tid >> 4, sm = tid & 15;
    const float* src = X + (size_t)b * bStride + (size_t)sn * nStride + (size_t)(t0 + sm) * tStride;
    float* dst = Xs + (sn * 16 + sm) * K;
    for (int f = 0; f < K; f += 4) async_load_b128(dst + f, src + f);
  }
  async_wait0();
  __syncthreads();

  const int ksteps = K >> 2;
  const int n0i = wave * 2, n1i = n0i + 1;
  const float* arow0 = Xs + (n0i * 16 + ll) * K;
  const float* arow1 = Xs + (n1i * 16 + ll) * K;

  for (int p = 0; p < 8; ++p) {
    int nt0 = 2 * p;
    v8f c00 = V8F_ZERO, c01 = V8F_ZERO, c10 = V8F_ZERO, c11 = V8F_ZERO;
    const float* bp0 = Wp + ((size_t)(nt0 * ksteps) * 32 + lane) * 2;
    const float* bp1 = bp0 + (size_t)ksteps * 64;
    for (int ks = 0; ks < ksteps; ++ks) {
      int kk = ks * 4 + 2 * lh;
      v2f a0 = *(const v2f*)(arow0 + kk);
      v2f a1 = *(const v2f*)(arow1 + kk);
      v2f b0 = *(const v2f*)(bp0 + ks * 64);
      v2f b1 = *(const v2f*)(bp1 + ks * 64);
      c00 = __builtin_amdgcn_wmma_f32_16x16x4_f32(false, a0, false, b0, (short)0, c00, false, false);
      c01 = __builtin_amdgcn_wmma_f32_16x16x4_f32(false, a0, false, b1, (short)0, c01, false, false);
      c10 = __builtin_amdgcn_wmma_f32_16x16x4_f32(false, a1, false, b0, (short)0, c10, false, false);
      c11 = __builtin_amdgcn_wmma_f32_16x16x4_f32(false, a1, false, b1, (short)0, c11, false, false);
    }
    // store 4 tiles (D layout: M = v + 8*laneHalf, N = lane&15)
    {
      float* d00 = (nt0 < 8 ? As : Ys) + (n0i * 16) * 128 + (nt0 & 7) * 16 + ll;
      float* d01 = ((nt0 + 1) < 8 ? As : Ys) + (n0i * 16) * 128 + ((nt0 + 1) & 7) * 16 + ll;
      float* d10 = (nt0 < 8 ? As : Ys) + (n1i * 16) * 128 + (nt0 & 7) * 16 + ll;
      float* d11 = ((nt0 + 1) < 8 ? As : Ys) + (n1i * 16) * 128 + ((nt0 + 1) & 7) * 16 + ll;
#pragma unroll
      for (int v = 0; v < 8; ++v) {
        int r = (v + 8 * lh) * 128;
        d00[r] = c00[v];
        d01[r] = c01[v];
        d10[r] = c10[v];
        d11[r] = c11[v];
      }
    }
  }
  __syncthreads();

  // --- cross-node max (excluding self) + bias + relu + store ---
  for (int it = tid; it < 16 * 128; it += blockDim.x) {
    int m = it >> 7, ch = it & 127;
    float m1 = NEG_INF, m2 = NEG_INF;
    int arg = -1;
#pragma unroll
    for (int n = 0; n < 8; ++n) {
      float y = Ys[(n * 16 + m) * 128 + ch];
      if (y > m1) { m2 = m1; m1 = y; arg = n; }
      else if (y > m2) { m2 = y; }
    }
    float bz = bias[ch];
#pragma unroll
    for (int n = 0; n < 8; ++n) {
      float o = As[(n * 16 + m) * 128 + ch] + bz + (n == arg ? m2 : m1);
      o = o > 0.f ? o : 0.f;
      S[((size_t)(b * 8 + n) * Tlen + (t0 + m)) * 132 + ch] = o;
    }
  }
  // --- context columns + zero pad ---
  if (tid < 128) {
    int n = tid >> 4, m = tid & 15;
    float* row = S + ((size_t)(b * 8 + n) * Tlen + (t0 + m)) * 132;
    const float* oc = ohs + (b * 8 + n) * 3;
    row[128] = oc[0];
    row[129] = oc[1];
    row[130] = oc[2];
    row[131] = 0.f;
  }
}

// ---------------------------------------------------------------------------
// Kernel 2/4: conv1d(k=3,same) + bias + relu + maxpool2.
//   1x4 register blocking: 4 WMMAs per {1 A-frag ds_b64 + 4 B-frag global_b64}.
//   Pool pairs live in the same lane of the D fragment (M = v + 8*laneHalf).
// ---------------------------------------------------------------------------
__global__ void conv_pool_kernel(const float* __restrict__ S,  // (256*Tlen) x 132
                                 int Tlen,
                                 const float* __restrict__ Wcp,  // packed [8][3][33][32][2]
                                 const float* __restrict__ bias, // 128
                                 float* __restrict__ Out) {      // (256*(Tlen/2)) x 128
  __shared__ float In[34 * 132];  // rows t0-1 .. t0+32

  const int bn = blockIdx.y;
  const int t0 = blockIdx.x * 32;
  const int tid = threadIdx.x;
  const int lane = tid & 31;
  const int wave = tid >> 5;
  const int lh = lane >> 4;
  const int ll = lane & 15;

  // async stage (34 rows x 33 float4), zero-fill halo outside the sequence
  for (int idx = tid; idx < 34 * 33; idx += blockDim.x) {
    int r = idx / 33, q = idx - r * 33;
    int t = t0 - 1 + r;
    float* dst = In + r * 132 + q * 4;
    if (t >= 0 && t < Tlen) {
      async_load_b128(dst, S + ((size_t)bn * Tlen + t) * 132 + q * 4);
    } else {
      dst[0] = 0.f; dst[1] = 0.f; dst[2] = 0.f; dst[3] = 0.f;
    }
  }
  async_wait0();
  __syncthreads();

  const int mt = wave >> 1;  // t sub-tile (rows mt*16..mt*16+15)
  const int nq = wave & 1;   // co half (ct = nq*4 .. nq*4+3)
  v8f c[4] = {V8F_ZERO, V8F_ZERO, V8F_ZERO, V8F_ZERO};

  for (int dk = 0; dk < 3; ++dk) {
    const float* a0 = In + (mt * 16 + dk + ll) * 132;
    for (int ks = 0; ks < 33; ++ks) {
      int kk = ks * 4 + 2 * lh;
      v2f a = *(const v2f*)(a0 + kk);
#pragma unroll
      for (int j = 0; j < 4; ++j) {
        int ct = nq * 4 + j;
        v2f bf = *(const v2f*)(Wcp + (((size_t)(ct * 3 + dk) * 33 + ks) * 32 + lane) * 2);
        c[j] = __builtin_amdgcn_wmma_f32_16x16x4_f32(false, a, false, bf, (short)0, c[j], false, false);
      }
    }
  }

#pragma unroll
  for (int j = 0; j < 4; ++j) {
    int co0 = nq * 64 + j * 16;
    float bz = bias[co0 + ll];
#pragma unroll
    for (int v = 0; v < 8; ++v) {
      float z = c[j][v] + bz;
      c[j][v] = z > 0.f ? z : 0.f;
    }
    int tpbase = (t0 + mt * 16) >> 1;
#pragma unroll
    for (int u = 0; u < 4; ++u) {
      float p = fmaxf(c[j][2 * u], c[j][2 * u + 1]);  // maxpool pair within lane
      int tp = tpbase + u + 4 * lh;
      Out[((size_t)bn * (Tlen >> 1) + tp) * 128 + co0 + ll] = p;
    }
  }
}

// ---------------------------------------------------------------------------
extern "C" void kernel_launch(void* const* d_in, const int* in_sizes, int n_in,
                              void* d_out, int out_size, void* d_ws, size_t ws_size,
                              hipStream_t stream) {
  const float* data = (const float*)d_in[0];   // (256,32,8,64)
  const float* ohs  = (const float*)d_in[1];   // (256,32,8,3); t=0 slice used
  const float* We0  = (const float*)d_in[4];   // (128,128)
  const float* be0  = (const float*)d_in[5];   // (128)
  const float* We1  = (const float*)d_in[6];   // (256,128)
  const float* be1  = (const float*)d_in[7];   // (128)
  const float* Wc0  = (const float*)d_in[8];   // (128,131,3)
  const float* bc0  = (const float*)d_in[9];   // (128)
  const float* Wc1  = (const float*)d_in[10];  // (128,131,3)
  const float* bc1  = (const float*)d_in[11];  // (128)
  (void)in_sizes; (void)n_in; (void)out_size; (void)ws_size;

  float* ws = (float*)d_ws;
  float* Wep0 = ws; ws += 16 * 16 * 64;         // 16384
  float* Wep1 = ws; ws += 16 * 32 * 64;         // 32768
  float* Wcp0 = ws; ws += 8 * 3 * 33 * 64;      // 50688
  float* Wcp1 = ws; ws += 8 * 3 * 33 * 64;      // 50688
  float* S0   = ws; ws += (size_t)65536 * 132;  // layer0 conv input
  float* X1   = ws; ws += (size_t)32768 * 128;  // pooled layer0 output
  float* S1   = ws; ws += (size_t)32768 * 132;  // layer1 conv input

  prep_weights<<<64, 256, 0, stream>>>(We0, We1, Wc0, Wc1, Wep0, Wep1, Wcp0, Wcp1);

  // EdgeConv layer 0: K=64, Tlen=256. data strides (floats): b=512, n=64, t=16384.
  {
    dim3 grid(16, 32);
    size_t shmem = (size_t)(128 * 64 + 16384 + 16384) * sizeof(float);  // 160 KB
    edge_conv_kernel<<<grid, 128, shmem, stream>>>(data, 8 * 64, 64, 32 * 8 * 64,
                                                   64, 256, Wep0, be0, ohs, S0);
  }
  conv_pool_kernel<<<dim3(8, 256), 128, 0, stream>>>(S0, 256, Wcp0, bc0, X1);

  // EdgeConv layer 1: K=128, Tlen=128. X1 strides: b=131072, n=16384, t=128.
  {
    dim3 grid(8, 32);
    size_t shmem = (size_t)(128 * 128 + 16384 + 16384) * sizeof(float);  // 192 KB
    edge_conv_kernel<<<grid, 128, shmem, stream>>>(X1, 8 * 128 * 128, 128 * 128, 128,
                                                   128, 128, Wep1, be1, ohs, S1);
  }
  conv_pool_kernel<<<dim3(4, 256), 128, 0, stream>>>(S1, 128, Wcp1, bc1, (float*)d_out);
}